// LCAM_62311385530669
// MI455X (gfx1250) — compile-verified
//
#include <hip/hip_runtime.h>
#include <math.h>

// Problem constants (from reference)
#define BATCH 16
#define CCH   512
#define HH    64
#define WW    64
#define MIPD  64      // MIP = max(8, C//RED)
#define LLEN  128     // H + W
#define EPSV  1e-5f

typedef __attribute__((ext_vector_type(2))) float v2f;
typedef __attribute__((ext_vector_type(4))) float f4v;
typedef __attribute__((ext_vector_type(8))) float v8f;

// ---------------------------------------------------------------------------
// One wave computes a 16x16 fp32 D tile of A(16xK,row-major) * B(Kx16,row-major)
// using V_WMMA_F32_16X16X4_F32.
// A frag layout (ISA 7.12.2, 32-bit A 16x4): lane<16 -> M=lane, K={k,k+1};
//   lane>=16 -> M=lane-16, K={k+2,k+3}.  B frag mirrors with N=lane%16.
// D layout: VGPR r -> M = r + 8*(lane>=16), N = lane%16.
// ---------------------------------------------------------------------------
template <int K>
__device__ inline v8f wmma_tile_f32(const float* __restrict__ A, int lda,
                                    const float* __restrict__ Bm, int ldb) {
  const int lane = threadIdx.x & 31;
  const int half = lane >> 4;
  const int mn   = lane & 15;
  v8f acc = {0.f, 0.f, 0.f, 0.f, 0.f, 0.f, 0.f, 0.f};
#if __has_builtin(__builtin_amdgcn_wmma_f32_16x16x4_f32)
  for (int k = 0; k < K; k += 4) {
    const int ka = k + half * 2;
    v2f a, b;
    a[0] = A[mn * lda + ka];
    a[1] = A[mn * lda + ka + 1];
    b[0] = Bm[ka * ldb + mn];
    b[1] = Bm[(ka + 1) * ldb + mn];
    acc = __builtin_amdgcn_wmma_f32_16x16x4_f32(false, a, false, b,
                                                (short)0, acc, false, false);
  }
#else
  // Scalar fallback (keeps file compiling if builtin missing; histogram will show wmma=0)
  for (int r = 0; r < 8; ++r) {
    const int m = r + half * 8;
    float s = 0.f;
    for (int k = 0; k < K; ++k) s += A[m * lda + k] * Bm[k * ldb + mn];
    acc[r] = s;
  }
#endif
  return acc;
}

// ---------------------------------------------------------------------------
// Kernel 1: per-(b,c) plane reductions. One block per plane (64x64 floats).
// Produces mean_HW, max_HW, row means (x_h), column means (x_w).
// Leaves x hot in the 192MB L2 for the final pass.
// ---------------------------------------------------------------------------
__global__ void k_reduce(const float* __restrict__ x,
                         float* __restrict__ redmean, float* __restrict__ redmax,
                         float* __restrict__ xh, float* __restrict__ xw) {
  const int plane = blockIdx.x;                   // b*C + c
  const float* p = x + (size_t)plane * 4096;
  __shared__ float tile[4096];
  __shared__ float rowsum[64];
  __shared__ float sbuf[8];
  __shared__ float mbuf[8];

  const int t = threadIdx.x;                      // 256 threads
  const int row = t >> 2, seg = t & 3;
  const float4* p4 = (const float4*)(p + row * 64 + seg * 16);

  float s = 0.f, mx = -INFINITY;
#pragma unroll
  for (int i = 0; i < 4; ++i) {
    float4 v = p4[i];
    ((float4*)tile)[t * 4 + i] = v;
    s += v.x + v.y + v.z + v.w;
    mx = fmaxf(mx, fmaxf(fmaxf(v.x, v.y), fmaxf(v.z, v.w)));
  }
  // row sum: 4 contiguous lanes per row (stays within a wave32)
  float rs = s;
  rs += __shfl_xor(rs, 1, 32);
  rs += __shfl_xor(rs, 2, 32);
  if (seg == 0) rowsum[row] = rs;
  // wave-level total sum & max
  float ts = s, tm = mx;
  for (int off = 16; off > 0; off >>= 1) {
    ts += __shfl_xor(ts, off, 32);
    tm = fmaxf(tm, __shfl_xor(tm, off, 32));
  }
  const int wid = t >> 5;
  if ((t & 31) == 0) { sbuf[wid] = ts; mbuf[wid] = tm; }
  __syncthreads();
  if (t == 0) {
    float S = 0.f, M = -INFINITY;
    for (int i = 0; i < 8; ++i) { S += sbuf[i]; M = fmaxf(M, mbuf[i]); }
    redmean[plane] = S * (1.f / 4096.f);
    redmax[plane]  = M;
  }
  if (t < 64) {
    xh[(size_t)plane * 64 + t] = rowsum[t] * (1.f / 64.f);
    // column sums from LDS tile: lane t always hits bank t -> conflict-free
    float cs = 0.f;
    for (int r = 0; r < 64; ++r) cs += tile[r * 64 + t];
    xw[(size_t)plane * 64 + t] = cs * (1.f / 64.f);
  }
}

// ---------------------------------------------------------------------------
// Kernel 2: channel-attention gate. One block per batch.
// gate1p[b][c] = 1 + sigmoid(w_conv @ [relu(w_conv1@max); relu(w_conv1@avg)] + b_conv)
// ---------------------------------------------------------------------------
__global__ void k_gate(const float* __restrict__ redmean, const float* __restrict__ redmax,
                       const float* __restrict__ w1, const float* __restrict__ b1,
                       const float* __restrict__ wg, const float* __restrict__ bg,
                       float* __restrict__ gate1p) {
  const int b = blockIdx.x;
  __shared__ float avgv[CCH], maxv[CCH], cc[2 * MIPD];
  const int t = threadIdx.x;
  for (int i = t; i < CCH; i += 256) {
    avgv[i] = redmean[b * CCH + i];
    maxv[i] = redmax[b * CCH + i];
  }
  __syncthreads();
  if (t < 2 * MIPD) {
    const int m = t & (MIPD - 1);
    const float* src = (t < MIPD) ? maxv : avgv;   // concat order: [x_max, x_avg]
    float s = b1[m];
    for (int c = 0; c < CCH; ++c) s += w1[m * CCH + c] * src[c];
    cc[t] = fmaxf(s, 0.f);
  }
  __syncthreads();
  for (int c = t; c < CCH; c += 256) {
    float s = bg[c];
    for (int j = 0; j < 2 * MIPD; ++j) s += wg[c * (2 * MIPD) + j] * cc[j];
    gate1p[b * CCH + c] = 1.f + 1.f / (1.f + expf(-s));
  }
}

// ---------------------------------------------------------------------------
// Kernel 3a: t1 = hswish(bn1(w_conv1 @ [x_h ; x_w] + b_conv1))   (B x 64 x 128)
// One wave per 16x16 tile. K = 512.
// ---------------------------------------------------------------------------
__global__ void k_t1(const float* __restrict__ xh, const float* __restrict__ xw,
                     const float* __restrict__ w1, const float* __restrict__ b1,
                     const float* __restrict__ bn_g, const float* __restrict__ bn_b,
                     const float* __restrict__ bn_m, const float* __restrict__ bn_v,
                     float* __restrict__ t1) {
  const int blk = blockIdx.x;
  const int b = blk >> 5;                 // 32 tiles per batch (4 M x 8 N)
  const int tile = blk & 31;
  const int m0 = (tile >> 3) * 16, n0 = (tile & 7) * 16;
  const float* Bm = (n0 < HH) ? (xh + (size_t)b * CCH * HH + n0)
                              : (xw + (size_t)b * CCH * WW + (n0 - HH));
  v8f acc = wmma_tile_f32<CCH>(w1 + m0 * CCH, CCH, Bm, HH);
  const int lane = threadIdx.x & 31;
  const int half = lane >> 4, mn = lane & 15;
  const int l = n0 + mn;
#pragma unroll
  for (int r = 0; r < 8; ++r) {
    const int m = m0 + r + half * 8;
    float val = acc[r] + b1[m];
    const float s = bn_g[m] * rsqrtf(bn_v[m] + EPSV);
    val = (val - bn_m[m]) * s + bn_b[m];
    val = val * fminf(fmaxf(val + 3.f, 0.f), 6.f) * (1.f / 6.f);   // hswish
    t1[(size_t)b * MIPD * LLEN + m * LLEN + l] = val;
  }
}

// ---------------------------------------------------------------------------
// Kernel 3b: y2 = w_conv0 @ t1 + b_conv0   (B x 512 x 128), K = 64
// ---------------------------------------------------------------------------
__global__ void k_y2(const float* __restrict__ t1, const float* __restrict__ w0,
                     const float* __restrict__ b0, float* __restrict__ y2) {
  const int blk = blockIdx.x;
  const int b = blk >> 8;                 // 256 tiles per batch (32 M x 8 N)
  const int tile = blk & 255;
  const int c0 = (tile >> 3) * 16, n0 = (tile & 7) * 16;
  v8f acc = wmma_tile_f32<MIPD>(w0 + c0 * MIPD, MIPD,
                                t1 + (size_t)b * MIPD * LLEN + n0, LLEN);
  const int lane = threadIdx.x & 31;
  const int half = lane >> 4, mn = lane & 15;
  const int l = n0 + mn;
#pragma unroll
  for (int r = 0; r < 8; ++r) {
    const int c = c0 + r + half * 8;
    y2[(size_t)b * CCH * LLEN + c * LLEN + l] = acc[r] + b0[c];
  }
}

// ---------------------------------------------------------------------------
// Kernel 3c: v = w_v @ y2 + b_v ; outhw = bn2(2*v) + y2   (B x 512 x 128), K = 512
// (softmax over singleton axis == 1, so att==1 and out = bn2(v+v)+residual)
// ---------------------------------------------------------------------------
__global__ void k_out(const float* __restrict__ y2, const float* __restrict__ wv,
                      const float* __restrict__ bv,
                      const float* __restrict__ bn_g, const float* __restrict__ bn_b,
                      const float* __restrict__ bn_m, const float* __restrict__ bn_v,
                      float* __restrict__ outhw) {
  const int blk = blockIdx.x;
  const int b = blk >> 8;
  const int tile = blk & 255;
  const int c0 = (tile >> 3) * 16, n0 = (tile & 7) * 16;
  const float* Bm = y2 + (size_t)b * CCH * LLEN + n0;
  v8f acc = wmma_tile_f32<CCH>(wv + c0 * CCH, CCH, Bm, LLEN);
  const int lane = threadIdx.x & 31;
  const int half = lane >> 4, mn = lane & 15;
  const int l = n0 + mn;
#pragma unroll
  for (int r = 0; r < 8; ++r) {
    const int c = c0 + r + half * 8;
    const float vv = acc[r] + bv[c];
    const float s = bn_g[c] * rsqrtf(bn_v[c] + EPSV);
    const float val = (2.f * vv - bn_m[c]) * s + bn_b[c]
                    + y2[(size_t)b * CCH * LLEN + c * LLEN + l];
    outhw[(size_t)b * CCH * LLEN + c * LLEN + l] = val;
  }
}

// ---------------------------------------------------------------------------
// Kernel 4: out[b,c,h,w] = x*(1+gate[b,c]) + out_h[b,c,h]*out_w[b,c,w]
// x reads should hit L2 (left resident by k_reduce); output stores are
// non-temporal so the 128MB output stream doesn't evict x from L2.
// ---------------------------------------------------------------------------
__global__ void k_final(const float* __restrict__ x, const float* __restrict__ gate1p,
                        const float* __restrict__ outhw, float* __restrict__ out) {
  const int plane = blockIdx.x;
  const int t = threadIdx.x;
  const float g = gate1p[plane];
  const float* oh = outhw + (size_t)plane * LLEN;   // [0,64)=h part, [64,128)=w part
  const int row = t >> 2, seg = t & 3;
  const float ohv = oh[row];
  const f4v* owp = (const f4v*)(oh + 64 + seg * 16);
  const size_t base = (size_t)plane * 4096 + row * 64 + seg * 16;
  const f4v* xp = (const f4v*)(x + base);
  f4v* op = (f4v*)(out + base);
#pragma unroll
  for (int i = 0; i < 4; ++i) {
    const f4v ow = owp[i];
    const f4v xv = xp[i];
    const f4v o = xv * g + ohv * ow;
    __builtin_nontemporal_store(o, op + i);
  }
}

// ---------------------------------------------------------------------------
extern "C" void kernel_launch(void* const* d_in, const int* in_sizes, int n_in,
                              void* d_out, int out_size, void* d_ws, size_t ws_size,
                              hipStream_t stream) {
  (void)in_sizes; (void)n_in; (void)out_size; (void)ws_size;
  const float* x       = (const float*)d_in[0];
  const float* w_conv1 = (const float*)d_in[1];
  const float* b_conv1 = (const float*)d_in[2];
  const float* w_conv  = (const float*)d_in[3];
  const float* b_conv  = (const float*)d_in[4];
  const float* bn1_g   = (const float*)d_in[5];
  const float* bn1_b   = (const float*)d_in[6];
  const float* bn1_m   = (const float*)d_in[7];
  const float* bn1_v   = (const float*)d_in[8];
  const float* w_conv0 = (const float*)d_in[9];
  const float* b_conv0 = (const float*)d_in[10];
  // d_in[11..14] (w_q,b_q,w_k,b_k) and d_in[17,18] (w_conv4,b_conv4) are
  // mathematically dead: softmax over a size-1 axis is exactly 1.0.
  const float* w_v     = (const float*)d_in[15];
  const float* b_v     = (const float*)d_in[16];
  const float* bn2_g   = (const float*)d_in[19];
  const float* bn2_b   = (const float*)d_in[20];
  const float* bn2_m   = (const float*)d_in[21];
  const float* bn2_v   = (const float*)d_in[22];
  float* out = (float*)d_out;

  // Workspace layout (floats); total ~3.30M floats (~12.6 MiB)
  float* ws      = (float*)d_ws;
  float* redmean = ws;                       //  8192
  float* redmax  = ws + 8192;                //  8192
  float* gate1p  = ws + 16384;               //  8192
  float* xh      = ws + 24576;               //  B*C*64 = 524288
  float* xw      = xh + 524288;              //  524288
  float* t1      = xw + 524288;              //  B*64*128 = 131072
  float* y2      = t1 + 131072;              //  B*512*128 = 1048576
  float* outhw   = y2 + 1048576;             //  1048576

  k_reduce<<<BATCH * CCH, 256, 0, stream>>>(x, redmean, redmax, xh, xw);
  k_gate  <<<BATCH,       256, 0, stream>>>(redmean, redmax, w_conv1, b_conv1,
                                            w_conv, b_conv, gate1p);
  k_t1    <<<BATCH * 32,   32, 0, stream>>>(xh, xw, w_conv1, b_conv1,
                                            bn1_g, bn1_b, bn1_m, bn1_v, t1);
  k_y2    <<<BATCH * 256,  32, 0, stream>>>(t1, w_conv0, b_conv0, y2);
  k_out   <<<BATCH * 256,  32, 0, stream>>>(y2, w_v, b_v,
                                            bn2_g, bn2_b, bn2_m, bn2_v, outhw);
  k_final <<<BATCH * CCH, 256, 0, stream>>>(x, gate1p, outhw, out);
}